// SpikeFP32VecSub_11450382811500
// MI455X (gfx1250) — compile-verified
//
#include <hip/hip_runtime.h>
#include <cstdint>
#include <cstddef>

// MI455X (gfx1250) spiking-FP32 subtract on IEEE-754 bit planes.
//
// Memory-bound: 768 MB/call -> ~33 us floor @ 23.3 TB/s. Strategy:
//  * async global->LDS staging (coalesced 512B per instruction),
//  * XOR-swizzled LDS layout: conflict-minimal DMA writes AND ds_read_b128,
//  * pack bits -> u32, hardware v_sub_f32 (bit-exact vs reference decode/add),
//  * reference encode edge cases: zero -> all-0 bits; subnormal -> sign|exp=1|mant=0,
//  * each thread stores one full contiguous 128B line (8x b128) -> full-line HBM WB.

#define GAS __attribute__((address_space(1)))
#define LAS __attribute__((address_space(3)))

typedef int v4i __attribute__((ext_vector_type(4)));

__device__ __forceinline__ void async_load_b128(const float* g, uint4* l) {
#if __has_builtin(__builtin_amdgcn_global_load_async_to_lds_b128)
  __builtin_amdgcn_global_load_async_to_lds_b128((GAS v4i*)g, (LAS v4i*)l, 0, 0);
#else
  unsigned loff = (unsigned)(size_t)(LAS char*)l;
  asm volatile("global_load_async_to_lds_b128 %0, %1, off"
               :: "v"(loff), "v"(g) : "memory");
#endif
}

__device__ __forceinline__ void async_wait0() {
#if __has_builtin(__builtin_amdgcn_s_wait_asynccnt)
  __builtin_amdgcn_s_wait_asynccnt(0);
#else
  asm volatile("s_wait_asynccnt 0" ::: "memory");
#endif
}

// Encode edge cases to match reference exactly.
__device__ __forceinline__ unsigned fix_encode(unsigned uc) {
  if ((uc & 0x7F800000u) == 0u) {              // zero or subnormal result
    uc = (uc & 0x007FFFFFu) ? ((uc & 0x80000000u) | 0x00800000u)  // subnorm -> +-2^-126 pattern
                            : 0u;                                  // +-0 -> all zeros
  }
  return uc;
}

__global__ __launch_bounds__(256) void spike_sub_tile(
    const float* __restrict__ A, const float* __restrict__ B,
    float* __restrict__ O) {
  // Tile: 256 elements x 128B (32 bit-plane floats) per matrix = 32KB each.
  __shared__ uint4 sA[2048];
  __shared__ uint4 sB[2048];

  const int t = threadIdx.x;
  const long long tileChunk = (long long)blockIdx.x * 2048;  // 16B chunks per tile
  const float* gA = A + tileChunk * 4;
  const float* gB = B + tileChunk * 4;

  // Cooperative async DMA: chunk c = t + 256k (consecutive lanes -> consecutive
  // 16B chunks -> coalesced 512B/instr). LDS slot is XOR-swizzled.
#pragma unroll
  for (int k = 0; k < 8; ++k) {
    int c    = t + 256 * k;
    int e    = c >> 3;
    int i    = c & 7;
    int slot = e * 8 + (i ^ ((e >> 1) & 7));
    async_load_b128(gA + c * 4, &sA[slot]);
    async_load_b128(gB + c * 4, &sB[slot]);
  }
  async_wait0();
  __syncthreads();

  // Pack: element t, logical chunk i lives at physical slot i ^ sw.
  const unsigned sw = ((unsigned)t >> 1) & 7u;
  unsigned ua = 0u, ub = 0u;
#pragma unroll
  for (int i = 0; i < 8; ++i) {
    uint4 qa = sA[t * 8 + (i ^ (int)sw)];
    uint4 qb = sB[t * 8 + (i ^ (int)sw)];
    // floats are exactly 0.0f (0x0) or 1.0f (0x3F800000): extract one bit each.
    unsigned na = ((qa.x >> 26) & 8u) | ((qa.y >> 27) & 4u) |
                  ((qa.z >> 28) & 2u) | ((qa.w >> 29) & 1u);
    unsigned nb = ((qb.x >> 26) & 8u) | ((qb.y >> 27) & 4u) |
                  ((qb.z >> 28) & 2u) | ((qb.w >> 29) & 1u);
    ua |= na << (28 - 4 * i);
    ub |= nb << (28 - 4 * i);
  }

  float    fc = __uint_as_float(ua) - __uint_as_float(ub);  // == A + (-B), RNE
  unsigned uc = fix_encode(__float_as_uint(fc));

  // Unpack: thread writes one contiguous 128B line.
  uint4* outp = (uint4*)O + ((size_t)blockIdx.x * 256 + (size_t)t) * 8;
#pragma unroll
  for (int p = 0; p < 8; ++p) {
    uint4 o;
    o.x = ((uc >> (31 - 4 * p    )) & 1u) ? 0x3F800000u : 0u;
    o.y = ((uc >> (31 - 4 * p - 1)) & 1u) ? 0x3F800000u : 0u;
    o.z = ((uc >> (31 - 4 * p - 2)) & 1u) ? 0x3F800000u : 0u;
    o.w = ((uc >> (31 - 4 * p - 3)) & 1u) ? 0x3F800000u : 0u;
    outp[p] = o;
  }
}

// Scalar tail (only launched if element count is not a multiple of 256).
__global__ void spike_sub_tail(const float* __restrict__ A,
                               const float* __restrict__ B,
                               float* __restrict__ O,
                               long long base, long long nElem) {
  long long e = base + (long long)blockIdx.x * blockDim.x + threadIdx.x;
  if (e >= nElem) return;
  const float* a = A + e * 32;
  const float* b = B + e * 32;
  unsigned ua = 0u, ub = 0u;
#pragma unroll
  for (int j = 0; j < 32; ++j) {
    ua |= (__float_as_uint(a[j]) ? 1u : 0u) << (31 - j);
    ub |= (__float_as_uint(b[j]) ? 1u : 0u) << (31 - j);
  }
  float    fc = __uint_as_float(ua) - __uint_as_float(ub);
  unsigned uc = fix_encode(__float_as_uint(fc));
  float* o = O + e * 32;
#pragma unroll
  for (int j = 0; j < 32; ++j)
    o[j] = ((uc >> (31 - j)) & 1u) ? 1.0f : 0.0f;
}

extern "C" void kernel_launch(void* const* d_in, const int* in_sizes, int n_in,
                              void* d_out, int out_size, void* d_ws, size_t ws_size,
                              hipStream_t stream) {
  const float* A = (const float*)d_in[0];
  const float* B = (const float*)d_in[1];
  float*       O = (float*)d_out;

  long long nElem  = (long long)out_size / 32;   // 2048*1024 = 2,097,152
  long long nTiles = nElem / 256;                // 8192 blocks
  if (nTiles > 0) {
    spike_sub_tile<<<dim3((unsigned)nTiles), dim3(256), 0, stream>>>(A, B, O);
  }
  long long rem = nElem - nTiles * 256;
  if (rem > 0) {
    spike_sub_tail<<<dim3((unsigned)((rem + 255) / 256)), dim3(256), 0, stream>>>(
        A, B, O, nTiles * 256, nElem);
  }
}